// MambaSSM_84447646974011
// MI455X (gfx1250) — compile-verified
//
#include <hip/hip_runtime.h>
#include <hip/hip_bf16.h>

// ---------------------------------------------------------------------------
// Mamba SSM forward for MI455X (gfx1250, wave32, WMMA)
//   1. in_proj GEMM (fp32 WMMA 16x16x4, pipelined) -> buf_x, buf_z
//   2. depthwise causal conv(4) + SiLU             -> buf_xc
//   3. x_proj GEMM (WMMA)                          -> buf_dbc [4096 x 80]
//   4. dt_proj GEMM + bias + softplus (WMMA)       -> buf_x (recycled: delta)
//   5. selective scan, lane-per-state, shuffle reduce, fused gating -> buf_y
//   6. out_proj GEMM (WMMA)                        -> d_out
// ---------------------------------------------------------------------------

#define D_MODEL 768
#define D_STATE 16
#define D_CONV  4
#define D_INNER 1536
#define DT_RANK 48
#define BATCH   2
#define SEQ     2048
#define MROWS   (BATCH * SEQ)            // 4096 token rows
#define DBC_LD  (DT_RANK + 2 * D_STATE)  // 80

typedef __attribute__((ext_vector_type(2))) float v2f;
typedef __attribute__((ext_vector_type(8))) float v8f;

__device__ __forceinline__ float silu_f(float x) {
    return x * (1.0f / (1.0f + __expf(-x)));
}
__device__ __forceinline__ float softplus_f(float x) {
    return (x > 20.0f) ? x : log1pf(__expf(x));
}

// ---------------------------------------------------------------------------
// C[M,N] = A[M,K] @ W[N,K]^T  (EPI==1: softplus(C + bias[n]))
// Wave computes (16*MT) x (16*NT) tile via V_WMMA_F32_16X16X4_F32.
// Two-stage ping-pong register pipeline: loads for k-step i+1 issue before
// the WMMAs of step i, so waits become loadcnt<=MT+NT instead of 0.
// A/B fragment lane map (ISA 7.12.2): row = lane&15, k-offset = 2*(lane>>4).
// EXEC all-1s: no divergence, exact tile multiples. K must be a multiple of 8.
// ---------------------------------------------------------------------------
template <int MT, int NT, int EPI>
__launch_bounds__(128)
__global__ void gemm_wmma_f32(const float* __restrict__ A,
                              const float* __restrict__ W,
                              const float* __restrict__ bias,
                              float* __restrict__ C,
                              int M, int N, int K, int lda, int ldc) {
    const int lane  = threadIdx.x & 31;
    const int wave  = threadIdx.x >> 5;
    const int mBase = (blockIdx.y * (blockDim.x >> 5) + wave) * (16 * MT);
    const int nBase = blockIdx.x * (16 * NT);
    const int r16   = lane & 15;          // row within a 16-tile
    const int kOff  = (lane >> 4) * 2;    // per-lane K offset (0 or 2)

    const float* aP[MT];
    const float* wP[NT];
#pragma unroll
    for (int i = 0; i < MT; ++i)
        aP[i] = A + (size_t)(mBase + i * 16 + r16) * lda + kOff;
#pragma unroll
    for (int t = 0; t < NT; ++t)
        wP[t] = W + (size_t)(nBase + t * 16 + r16) * K + kOff;

    v8f acc[MT][NT];
#pragma unroll
    for (int i = 0; i < MT; ++i)
#pragma unroll
        for (int t = 0; t < NT; ++t) {
            v8f z = {0.f, 0.f, 0.f, 0.f, 0.f, 0.f, 0.f, 0.f};
            acc[i][t] = z;
        }

    v2f aA[MT], bA[NT], aB[MT], bB[NT];

#define LOADFRAG(AF, BF, KK)                                         \
    _Pragma("unroll") for (int i = 0; i < MT; ++i)                   \
        AF[i] = *(const v2f*)(aP[i] + (KK));                         \
    _Pragma("unroll") for (int t = 0; t < NT; ++t)                   \
        BF[t] = *(const v2f*)(wP[t] + (KK));

#define MACFRAG(AF, BF)                                              \
    _Pragma("unroll") for (int i = 0; i < MT; ++i)                   \
        _Pragma("unroll") for (int t = 0; t < NT; ++t)               \
            acc[i][t] = __builtin_amdgcn_wmma_f32_16x16x4_f32(       \
                false, AF[i], false, BF[t], (short)0, acc[i][t],     \
                false, false);

    LOADFRAG(aA, bA, 0)
    int k = 0;
    for (; k + 8 < K; k += 8) {
        LOADFRAG(aB, bB, k + 4)   // prefetch step i+1
        MACFRAG(aA, bA)           // consume step i
        LOADFRAG(aA, bA, k + 8)   // prefetch step i+2
        MACFRAG(aB, bB)           // consume step i+1
    }
    // pipeline drain: steps at k and k+4 remain (K/4 is even)
    LOADFRAG(aB, bB, k + 4)
    MACFRAG(aA, bA)
    MACFRAG(aB, bB)

#undef LOADFRAG
#undef MACFRAG

    // C/D layout: VGPR r -> row (r + 8*(lane>>4)), col = lane&15 (ISA 7.12.2)
    const int rowSel = (lane >> 4) * 8;
#pragma unroll
    for (int i = 0; i < MT; ++i)
#pragma unroll
        for (int t = 0; t < NT; ++t) {
            const int col = nBase + t * 16 + (lane & 15);
            float bv = 0.0f;
            if (EPI == 1) bv = bias[col];
#pragma unroll
            for (int r = 0; r < 8; ++r) {
                float v = acc[i][t][r];
                if (EPI == 1) v = softplus_f(v + bv);
                C[(size_t)(mBase + i * 16 + r + rowSel) * ldc + col] = v;
            }
        }
}

// ---------------------------------------------------------------------------
// Depthwise causal conv (width 4) + bias + SiLU over [B,S,D_INNER]
// ---------------------------------------------------------------------------
__launch_bounds__(256)
__global__ void conv_silu_kernel(const float* __restrict__ xin,
                                 const float* __restrict__ w,
                                 const float* __restrict__ b,
                                 float* __restrict__ out) {
    const int idx = blockIdx.x * blockDim.x + threadIdx.x;  // B*S*D_INNER
    const int d  = idx % D_INNER;
    const int s  = (idx / D_INNER) % SEQ;
    const int bi = idx / (D_INNER * SEQ);

    const float* base = xin + (size_t)bi * SEQ * D_INNER + d;
    float acc = b[d];
#pragma unroll
    for (int j = 0; j < D_CONV; ++j) {
        const int sp = s - (D_CONV - 1) + j;
        if (sp >= 0) acc += w[d * D_CONV + j] * base[(size_t)sp * D_INNER];
    }
    out[idx] = silu_f(acc);
}

// ---------------------------------------------------------------------------
// Selective scan: one lane per (b, d, n).  h kept in a register, y reduced
// across the 16 state lanes with wave32 xor-shuffles, gating fused at write.
// 49152 lanes = 1536 waves of sequential-S work (latency-critical stage).
// ---------------------------------------------------------------------------
__launch_bounds__(256)
__global__ void scan_kernel(const float* __restrict__ delta,
                            const float* __restrict__ xc,
                            const float* __restrict__ dbc,
                            const float* __restrict__ zbuf,
                            const float* __restrict__ logA,
                            const float* __restrict__ Dskip,
                            float* __restrict__ y) {
    const int gid = blockIdx.x * blockDim.x + threadIdx.x;  // B*D_INNER*16
    const int n = gid & (D_STATE - 1);
    const int d = (gid >> 4) % D_INNER;
    const int b = gid / (D_INNER * D_STATE);

    const float Aneg   = -__expf(logA[d * D_STATE + n]);
    const bool  writer = (n == 0);
    const float dsk    = Dskip[d];

    const size_t baseBD  = (size_t)b * SEQ * D_INNER + d;
    const size_t baseDbc = (size_t)b * SEQ * DBC_LD;

    float h = 0.0f;
#pragma unroll 2
    for (int s = 0; s < SEQ; ++s) {
        const size_t ix = baseBD + (size_t)s * D_INNER;
        const float dt = delta[ix];     // broadcast across the 16 state lanes
        const float xv = xc[ix];
        const float* pd = dbc + baseDbc + (size_t)s * DBC_LD;
        const float Bv = pd[DT_RANK + n];
        const float Cv = pd[DT_RANK + D_STATE + n];

        const float a = __expf(dt * Aneg);
        h = a * h + (dt * Bv) * xv;

        float p = h * Cv;               // y = sum_n h*C : butterfly over n
        p += __shfl_xor(p, 1, 16);
        p += __shfl_xor(p, 2, 16);
        p += __shfl_xor(p, 4, 16);
        p += __shfl_xor(p, 8, 16);

        if (writer) {
            const float zv = zbuf[ix];
            y[ix] = (p + dsk * xv) * silu_f(zv);
        }
    }
}

// ---------------------------------------------------------------------------
extern "C" void kernel_launch(void* const* d_in, const int* in_sizes, int n_in,
                              void* d_out, int out_size, void* d_ws, size_t ws_size,
                              hipStream_t stream) {
    const float* x         = (const float*)d_in[0];
    const float* in_proj_w = (const float*)d_in[1];
    const float* conv_w    = (const float*)d_in[2];
    const float* conv_b    = (const float*)d_in[3];
    const float* x_proj_w  = (const float*)d_in[4];
    const float* dt_proj_w = (const float*)d_in[5];
    const float* dt_proj_b = (const float*)d_in[6];
    const float* log_A     = (const float*)d_in[7];
    const float* D_skip    = (const float*)d_in[8];
    const float* out_proj_w= (const float*)d_in[9];
    float* out = (float*)d_out;

    const size_t planeBD = (size_t)MROWS * D_INNER;   // 6,291,456 floats
    float* ws      = (float*)d_ws;
    float* buf_x   = ws;                              // raw x-branch -> delta
    float* buf_z   = buf_x  + planeBD;                // z-branch
    float* buf_xc  = buf_z  + planeBD;                // conv+silu output
    float* buf_dbc = buf_xc + planeBD;                // [4096 x 80]
    float* buf_y   = buf_dbc + (size_t)MROWS * DBC_LD;// gated scan output

    const dim3 blk(128);  // 4 waves; waves share the weight strip
    // wave tile = 32 rows x 64 cols (MT=2, NT=4); block covers 128 rows
    const int gy = MROWS / (16 * 2 * 4);  // 32

    // 1. in_proj: split into x-branch and z-branch GEMMs (N=1536 each, K=768)
    gemm_wmma_f32<2, 4, 0><<<dim3(D_INNER / 64, gy), blk, 0, stream>>>(
        x, in_proj_w, nullptr, buf_x, MROWS, D_INNER, D_MODEL, D_MODEL, D_INNER);
    gemm_wmma_f32<2, 4, 0><<<dim3(D_INNER / 64, gy), blk, 0, stream>>>(
        x, in_proj_w + (size_t)D_INNER * D_MODEL, nullptr, buf_z,
        MROWS, D_INNER, D_MODEL, D_MODEL, D_INNER);

    // 2. depthwise conv + SiLU
    conv_silu_kernel<<<(MROWS * D_INNER) / 256, 256, 0, stream>>>(
        buf_x, conv_w, conv_b, buf_xc);

    // 3. x_proj: N=80 (5 tiles of 16), K=1536
    gemm_wmma_f32<2, 1, 0><<<dim3(DBC_LD / 16, gy), blk, 0, stream>>>(
        buf_xc, x_proj_w, nullptr, buf_dbc, MROWS, DBC_LD, D_INNER, D_INNER, DBC_LD);

    // 4. dt_proj + bias + softplus: A = delta_raw (lda=80, K=48), recycle buf_x
    gemm_wmma_f32<2, 4, 1><<<dim3(D_INNER / 64, gy), blk, 0, stream>>>(
        buf_dbc, dt_proj_w, dt_proj_b, buf_x, MROWS, D_INNER, DT_RANK, DBC_LD, D_INNER);

    // 5. selective scan (+ fused D-skip and z-gating)
    scan_kernel<<<(BATCH * D_INNER * D_STATE) / 256, 256, 0, stream>>>(
        buf_x, buf_xc, buf_dbc, buf_z, log_A, D_skip, buf_y);

    // 6. out_proj: N=768, K=1536
    gemm_wmma_f32<2, 4, 0><<<dim3(D_MODEL / 64, gy), blk, 0, stream>>>(
        buf_y, out_proj_w, nullptr, out, MROWS, D_MODEL, D_INNER, D_INNER, D_MODEL);
}